// RegionLoss_31705448579680
// MI455X (gfx1250) — compile-verified
//
#include <hip/hip_runtime.h>
#include <math.h>

// Problem constants (fixed by the reference setup)
#define NB     64
#define NA_    5
#define NH     19
#define NW     19
#define NPIX   (NH*NW)        // 361
#define NANCH  (NA_*NPIX)     // 1805
#define NCHAN  (NA_*26)       // 130 channels in output tensor
#define NCLS   13
#define NOBJ   50
#define OBJP   64             // objects padded to 4x16 WMMA tiles
#define NATILE ((NANCH+15)/16) // 113
#define TH_    80.0f
#define SH_    2.0f
#define IMW_   640.0f
#define IMH_   480.0f
#define OBJ_SCALE 5.0f
#define SIL    0.4f
// exp(SH)-1 and SH/TH folded at compile time (avoid runtime libm + divide)
#define INV_CONF0 0.15651764274966565f   // 1/(e^2 - 1)
#define SH_OVER_TH 0.025f                // SH_/TH_

typedef __attribute__((ext_vector_type(2))) float v2f;
typedef __attribute__((ext_vector_type(8))) float v8f;

__device__ __forceinline__ float sigf(float x){ return 1.0f/(1.0f+__expf(-x)); }

// ---------------- K0: init scratch that must start at zero ----------------
__global__ void k_init(unsigned int* curconf, float* lossOut){
  int t = blockIdx.x*blockDim.x + threadIdx.x;
  if (t < NB*NANCH) curconf[t] = 0u;
  if (t == 0) lossOut[0] = 0.0f;
}

// ---------------- K1: per-anchor predicted corners (scaled) ----------------
__global__ void k_pred(const float* __restrict__ out, float* __restrict__ predsc,
                       float* __restrict__ pn, float* __restrict__ confsig){
  int t = blockIdx.x*blockDim.x + threadIdx.x;
  if (t >= NB*NANCH) return;
  int b = t / NANCH, anchor = t % NANCH;
  int a = anchor / NPIX, pix = anchor % NPIX;
  int h = pix / NW, w = pix % NW;
  const float* base = out + ((size_t)(b*NCHAN + a*26))*NPIX + pix;
  float x0 = sigf(base[0]);
  float y0 = sigf(base[(size_t)NPIX]);
  float px = (x0 + (float)w) * (1.0f/(float)NW);
  float py = (y0 + (float)h) * (1.0f/(float)NH);
  float P[12];
  P[0] = px*IMW_; P[1] = py*IMH_;
  #pragma unroll
  for (int k=0;k<5;++k){
    float rx = base[(size_t)(2+2*k)*NPIX];
    float ry = base[(size_t)(3+2*k)*NPIX];
    P[2+2*k] = (rx*(1.0f/(float)NW) + px)*IMW_;
    P[3+2*k] = (ry*(1.0f/(float)NH) + py)*IMH_;
  }
  float* pd = predsc + (size_t)t*12;
  float* pq = pn     + (size_t)t*6;
  #pragma unroll
  for (int j=0;j<12;++j) pd[j] = P[j];
  #pragma unroll
  for (int k=0;k<6;++k)  pq[k] = P[2*k]*P[2*k] + P[2*k+1]*P[2*k+1];
  confsig[t] = sigf(base[(size_t)12*NPIX]);
}

// ---------------- K2: per-target prep (sequential cumprod validity) -------
__global__ void k_tgt(const float* __restrict__ tgt, float* __restrict__ gsc,
                      float* __restrict__ gn, int* __restrict__ validA,
                      int* __restrict__ cellA, float* __restrict__ tvals,
                      int* __restrict__ tclsA){
  int b = blockIdx.x*blockDim.x + threadIdx.x;
  if (b >= NB) return;
  int v = 1;
  for (int o=0;o<OBJP;++o){
    int base = b*OBJP + o;
    if (o < NOBJ){
      const float* tp = tgt + (size_t)b*(NOBJ*15) + (size_t)o*15;
      float cls = tp[0];
      float g[12];
      #pragma unroll
      for (int j=0;j<12;++j) g[j] = tp[1+j];
      v = v && (g[0] != 0.0f);
      #pragma unroll
      for (int k=0;k<6;++k){
        float GX = g[2*k]*IMW_, GY = g[2*k+1]*IMH_;
        gsc[(size_t)base*12+2*k]   = GX;
        gsc[(size_t)base*12+2*k+1] = GY;
        gn[(size_t)base*6+k] = GX*GX + GY*GY;
      }
      int gi = (int)floorf(g[0]*(float)NW);
      int gj = (int)floorf(g[1]*(float)NH);
      validA[base] = v;
      cellA[base]  = v ? (gj*NW + gi) : -1;
      float tv[12];
      tv[0] = g[0]*(float)NW - (float)gi;
      tv[1] = g[1]*(float)NH - (float)gj;
      #pragma unroll
      for (int k=1;k<6;++k){
        tv[2*k]   = (g[2*k]  -g[0])*(float)NW;
        tv[2*k+1] = (g[2*k+1]-g[1])*(float)NH;
      }
      #pragma unroll
      for (int j=0;j<12;++j) tvals[(size_t)base*12+j] = tv[j];
      tclsA[base] = (int)cls;
    } else {
      validA[base] = 0; cellA[base] = -1; tclsA[base] = 0;
      #pragma unroll
      for (int j=0;j<12;++j){ gsc[(size_t)base*12+j]=0.f; tvals[(size_t)base*12+j]=0.f; }
      #pragma unroll
      for (int k=0;k<6;++k) gn[(size_t)base*6+k]=0.f;
    }
  }
}

// ---------------- K3: confs via WMMA f32 16x16x4 (cross terms) -----------
// dist_k^2 = |g_k|^2 + |p_k|^2 - 2 g_k.p_k ; cross terms are rank-2 GEMM tiles.
__global__ void __launch_bounds__(32)
k_confs(const float* __restrict__ predsc, const float* __restrict__ pn,
        const float* __restrict__ gsc, const float* __restrict__ gn,
        const int* __restrict__ validA, const int* __restrict__ cellA,
        unsigned int* __restrict__ curconf, float* __restrict__ tconfv){
  const int lane = threadIdx.x;
  const int b  = blockIdx.z;
  const int ot = blockIdx.y;      // object tile (4 tiles of 16 -> 64 padded)
  const int at = blockIdx.x;      // anchor tile (113 tiles of 16)
  const int objBase  = ot*16;
  const int anchBase = at*16;
  const int l16 = lane & 15;
  const bool hi = lane >= 16;
  const int anchor = anchBase + l16;
  const bool aok = anchor < NANCH;

  __shared__ float sgn[16*6];
  __shared__ int   sval[16];
  __shared__ int   scell[16];
  if (lane < 16){
    int obj = objBase + lane;
    #pragma unroll
    for (int k=0;k<6;++k) sgn[lane*6+k] = gn[(size_t)(b*OBJP+obj)*6+k];
    sval[lane]  = validA[b*OBJP+obj];
    scell[lane] = cellA[b*OBJP+obj];
  }
  __syncthreads();

  v2f A[6], Bv[6];
  v2f z2 = {0.0f, 0.0f};
  #pragma unroll
  for (int k=0;k<6;++k){ A[k]=z2; Bv[k]=z2; }
  if (!hi){
    // A: lanes 0-15 hold K=0,1 (gx,gy); lanes 16-31 are K=2,3 -> zero pad.
    const float* gp = gsc + (size_t)(b*OBJP + objBase + l16)*12;
    #pragma unroll
    for (int k=0;k<6;++k){ A[k].x = gp[2*k]; A[k].y = gp[2*k+1]; }
    if (aok){
      const float* pp = predsc + (size_t)(b*NANCH + anchor)*12;
      #pragma unroll
      for (int k=0;k<6;++k){ Bv[k].x = pp[2*k]; Bv[k].y = pp[2*k+1]; }
    }
  }

  v8f acc[6];
  #pragma unroll
  for (int k=0;k<6;++k){
    v8f c = {};
    acc[k] = __builtin_amdgcn_wmma_f32_16x16x4_f32(
        /*neg_a=*/false, A[k], /*neg_b=*/false, Bv[k],
        /*c_mod=*/(short)0, c, /*reuse_a=*/false, /*reuse_b=*/false);
  }

  float pl[6];
  if (aok){
    const float* pq = pn + (size_t)(b*NANCH + anchor)*6;
    #pragma unroll
    for (int k=0;k<6;++k) pl[k] = pq[k];
  } else {
    #pragma unroll
    for (int k=0;k<6;++k) pl[k] = 0.f;
  }

  #pragma unroll
  for (int r=0;r<8;++r){
    int M   = r + (hi ? 8 : 0);       // C/D layout: M = vgpr + 8*(lane>=16)
    int obj = objBase + M;
    float s = 0.f;
    #pragma unroll
    for (int k=0;k<6;++k){
      float d2 = sgn[M*6+k] + pl[k] - 2.0f*acc[k][r];
      d2 = d2 > 0.f ? d2 : 0.f;
      float dist = __builtin_amdgcn_sqrtf(d2);           // raw v_sqrt_f32
      // SH*(1 - dist/TH) == SH - dist*(SH/TH): one FMA, no divide
      float cpt = (dist < TH_) ? (__expf(SH_ - dist*SH_OVER_TH) - 1.0f)*INV_CONF0 : 0.f;
      s += cpt;
    }
    float conf = sval[M] ? (s*(1.0f/6.0f)) : 0.f;   // confs >= 0 always
    if (aok){
      atomicMax(&curconf[b*NANCH + anchor], __float_as_uint(conf)); // max over objs
      if (scell[M] == anchor) tconfv[b*OBJP + obj] = conf;          // tconf_v gather
    }
  }
}

// ---------------- K4: last-valid-writer per (b, cell) ---------------------
__global__ void k_winner(const int* __restrict__ validA, const int* __restrict__ cellA,
                         int* __restrict__ winner){
  int t = blockIdx.x*blockDim.x + threadIdx.x;
  if (t >= NB*NPIX) return;
  int b = t / NPIX, pix = t % NPIX;
  int w = -1;
  for (int o=0;o<NOBJ;++o){
    int base = b*OBJP + o;
    if (validA[base] && cellA[base] == pix) w = o;  // scan: last writer wins
  }
  winner[t] = w;
}

// ---------------- K5: loss terms + reduction ------------------------------
__global__ void k_loss(const float* __restrict__ out, const float* __restrict__ confsig,
                       const unsigned int* __restrict__ curconf, const int* __restrict__ winner,
                       const float* __restrict__ tconfv, const float* __restrict__ tvals,
                       const int* __restrict__ tclsA, float* __restrict__ lossOut){
  __shared__ float red[256];
  int t = blockIdx.x*blockDim.x + threadIdx.x;
  float sum = 0.f;
  if (t < NB*NANCH){
    int b = t / NANCH, anchor = t % NANCH;
    float conf  = confsig[t];
    float cur   = __uint_as_float(curconf[t]);
    float mask  = (cur < SIL) ? 1.0f : 0.0f;
    float tconf = 0.f;
    if (anchor < NPIX){                 // scan writes only anchor slot 0
      int w = winner[b*NPIX + anchor];
      if (w >= 0){
        mask  = OBJ_SCALE;
        tconf = tconfv[b*OBJP + w];
        const float* base = out + (size_t)(b*NCHAN)*NPIX + anchor;  // a = 0
        const float* tv   = tvals + (size_t)(b*OBJP + w)*12;
        // coord loss (COORD_SCALE == 1)
        float lc = 0.f;
        float p0 = sigf(base[0]);             { float d=p0-tv[0]; lc += d*d; }
        float p1 = sigf(base[(size_t)NPIX]);  { float d=p1-tv[1]; lc += d*d; }
        #pragma unroll
        for (int j=2;j<12;++j){ float p = base[(size_t)j*NPIX]; float d=p-tv[j]; lc += d*d; }
        sum += 0.5f*lc;
        // class NLL via log-softmax over 13 logits (channels 13..25)
        float l[NCLS]; float m = -3.0e38f;
        #pragma unroll
        for (int c=0;c<NCLS;++c){ l[c] = base[(size_t)(13+c)*NPIX]; m = l[c] > m ? l[c] : m; }
        float se = 0.f;
        #pragma unroll
        for (int c=0;c<NCLS;++c) se += __expf(l[c]-m);
        float lse = m + __logf(se);
        int cl = tclsA[b*OBJP + w];
        sum += (lse - l[cl]);           // CLASS_SCALE == 1
      }
    }
    float dc = conf - tconf;
    sum += 0.5f*dc*dc*mask;             // ((conf-tconf)*sqrt(mask))^2 == (.)^2*mask
  }
  red[threadIdx.x] = sum;
  __syncthreads();
  for (int s=128;s>0;s>>=1){
    if (threadIdx.x < s) red[threadIdx.x] += red[threadIdx.x+s];
    __syncthreads();
  }
  if (threadIdx.x == 0) atomicAdd(lossOut, red[0]);
}

// ---------------- host-side launch ----------------------------------------
extern "C" void kernel_launch(void* const* d_in, const int* in_sizes, int n_in,
                              void* d_out, int out_size, void* d_ws, size_t ws_size,
                              hipStream_t stream){
  const float* out = (const float*)d_in[0];   // (64, 130, 19, 19)
  const float* tgt = (const float*)d_in[1];   // (64, 750)
  float* loss = (float*)d_out;                // scalar

  char* ws = (char*)d_ws;
  size_t off = 0;
  auto carve = [&](size_t bytes)->char*{
    char* p = ws + off;
    off = (off + bytes + 255) & ~(size_t)255;
    return p;
  };
  float*        predsc  = (float*)       carve((size_t)NB*NANCH*12*sizeof(float)); // 5.5 MB
  float*        pn      = (float*)       carve((size_t)NB*NANCH*6 *sizeof(float)); // 2.8 MB
  float*        confsig = (float*)       carve((size_t)NB*NANCH   *sizeof(float));
  unsigned int* curconf = (unsigned int*)carve((size_t)NB*NANCH   *sizeof(unsigned int));
  float*        gsc     = (float*)       carve((size_t)NB*OBJP*12 *sizeof(float));
  float*        gn      = (float*)       carve((size_t)NB*OBJP*6  *sizeof(float));
  int*          validA  = (int*)         carve((size_t)NB*OBJP    *sizeof(int));
  int*          cellA   = (int*)         carve((size_t)NB*OBJP    *sizeof(int));
  float*        tvals   = (float*)       carve((size_t)NB*OBJP*12 *sizeof(float));
  int*          tclsA   = (int*)         carve((size_t)NB*OBJP    *sizeof(int));
  float*        tconfv  = (float*)       carve((size_t)NB*OBJP    *sizeof(float));
  int*          winner  = (int*)         carve((size_t)NB*NPIX    *sizeof(int));
  (void)ws_size; (void)in_sizes; (void)n_in; (void)out_size;

  const int n1 = NB*NANCH;
  k_init  <<<(n1+255)/256, 256, 0, stream>>>(curconf, loss);
  k_pred  <<<(n1+255)/256, 256, 0, stream>>>(out, predsc, pn, confsig);
  k_tgt   <<<1, NB, 0, stream>>>(tgt, gsc, gn, validA, cellA, tvals, tclsA);
  k_confs <<<dim3(NATILE, OBJP/16, NB), 32, 0, stream>>>(predsc, pn, gsc, gn,
                                                         validA, cellA, curconf, tconfv);
  k_winner<<<(NB*NPIX+255)/256, 256, 0, stream>>>(validA, cellA, winner);
  k_loss  <<<(n1+255)/256, 256, 0, stream>>>(out, confsig, curconf, winner,
                                             tconfv, tvals, tclsA, loss);
}